// gru_h_65266323030681
// MI455X (gfx1250) — compile-verified
//
#include <hip/hip_runtime.h>
#include <stdint.h>

// GRU stack, hidden_size=1, L=12 layers, T=12 steps, 7 passes, 65536 independent
// scalar streams. Compute-bound on transcendental chains; WMMA not applicable
// (rank-1 gates). Uses CDNA5 TDM (tensor_load_to_lds) to stage each block's
// contiguous 12KB input tile into LDS, then pure register/VALU recurrence.

#define LNUM   12
#define TNUM   12
#define NPASS  7          // 1 + HORIZON
#define BLK    256

typedef uint32_t u32x4 __attribute__((ext_vector_type(4)));
typedef uint32_t u32x8 __attribute__((ext_vector_type(8)));

__device__ __forceinline__ float fexp2(float x) { return __builtin_amdgcn_exp2f(x); }
__device__ __forceinline__ float frcp(float x)  { return __builtin_amdgcn_rcpf(x); }

// sigmoid(x) = 1/(1+2^(-x*log2e))   -> v_exp_f32 + v_rcp_f32
__device__ __forceinline__ float fsig(float x) {
  return frcp(1.0f + fexp2(-1.4426950408889634f * x));
}
// tanh(x) = 1 - 2/(2^(2x*log2e)+1)  (saturates correctly at +-1 on overflow)
__device__ __forceinline__ float ftanh_(float x) {
  return 1.0f - 2.0f * frcp(1.0f + fexp2(2.8853900817779268f * x));
}

__global__ __launch_bounds__(BLK) void gru_stack_kernel(
    const float* __restrict__ x,
    const float* __restrict__ w_ih, const float* __restrict__ w_hh,
    const float* __restrict__ b_ih, const float* __restrict__ b_hh,
    float* __restrict__ out, int E) {

  __shared__ float xs[BLK * TNUM];          // 12 KB tile: BLK elements x 12 steps

  const int tid = threadIdx.x;
  const int e   = blockIdx.x * BLK + tid;   // element id (b*N + n)

  // ---- TDM: DMA this block's contiguous 12KB slice of x into LDS ----------
  // x element (b,n) has its T=12 inputs contiguous (layout [B,1,N,T]), so a
  // block's inputs are one contiguous 3072-dword region starting at
  // x + blockIdx*3072. 1D tile: tile_dim0 = tensor_dim0 = 3072, dim1 = 1.
  if (tid < 32) {                           // one wave issues the DMA
    uint64_t gaddr = (uint64_t)(uintptr_t)x +
                     (uint64_t)blockIdx.x * (uint64_t)(BLK * TNUM * 4);
    uint32_t lds   = (uint32_t)(uintptr_t)(&xs[0]);  // low 32 bits = LDS offset
    uint32_t galo  = (uint32_t)__builtin_amdgcn_readfirstlane((int)(uint32_t)gaddr);
    uint32_t gahi  = (uint32_t)__builtin_amdgcn_readfirstlane((int)(uint32_t)(gaddr >> 32));
    lds            = (uint32_t)__builtin_amdgcn_readfirstlane((int)lds);

    const uint32_t nd = BLK * TNUM;         // 3072 dwords
    u32x4 g0;                               // D# group 0 (ISA 08 §8.3)
    g0[0] = 1u;                             // count=1 (valid), not restore
    g0[1] = lds;                            // lds_addr [63:32]
    g0[2] = galo;                           // global_addr[31:0]
    g0[3] = (gahi & 0x1FFFFFFu) | (2u << 30); // global_addr[56:32] | type=2

    u32x8 g1;                               // D# group 1 (ISA 08 §8.4)
    g1[0] = (2u << 16);                     // data_size=2 (4B); mask/flags=0
    g1[1] = (nd & 0xFFFFu) << 16;           // tensor_dim0[15:0] in [63:48]
    g1[2] = ((nd >> 16) & 0xFFFFu) | (1u << 16); // tensor_dim0[31:16] | tensor_dim1=1
    g1[3] = (nd & 0xFFFFu) << 16;           // tile_dim0 = 3072 in [127:112]
    g1[4] = 1u;                             // tile_dim1=1, tile_dim2=0
    g1[5] = nd;                             // tensor_dim0_stride[31:0]
    g1[6] = 0u;                             // stride hi / dim1_stride lo
    g1[7] = 0u;

    asm volatile("tensor_load_to_lds %0, %1" :: "s"(g0), "s"(g1) : "memory");
    __builtin_amdgcn_s_wait_tensorcnt(0);
  }
  __syncthreads();

  // ---- weights: wave-uniform scalar loads, biases pre-combined ------------
  float wi0[LNUM], wi1[LNUM], wi2[LNUM], wh0[LNUM], wh1[LNUM], wh2[LNUM];
  float b0[LNUM], b1[LNUM], bi2[LNUM], bh2[LNUM];
#pragma unroll
  for (int l = 0; l < LNUM; ++l) {
    wi0[l] = w_ih[3*l+0]; wi1[l] = w_ih[3*l+1]; wi2[l] = w_ih[3*l+2];
    wh0[l] = w_hh[3*l+0]; wh1[l] = w_hh[3*l+1]; wh2[l] = w_hh[3*l+2];
    b0[l]  = b_ih[3*l+0] + b_hh[3*l+0];
    b1[l]  = b_ih[3*l+1] + b_hh[3*l+1];
    bi2[l] = b_ih[3*l+2]; bh2[l] = b_hh[3*l+2];
  }

  if (e < E) {
    float h[LNUM];
#pragma unroll
    for (int l = 0; l < LNUM; ++l) h[l] = 0.0f;

    float* my = &xs[tid * TNUM];            // this thread's 12 sequence slots

    for (int p = 0; p < NPASS; ++p) {
      for (int t = 0; t < TNUM; ++t) {
        float inp = my[t];                  // ds_load_b32
#pragma unroll
        for (int l = 0; l < LNUM; ++l) {
          float hl  = h[l];
          float r   = fsig (fmaf(inp, wi0[l], fmaf(hl, wh0[l], b0[l])));
          float z   = fsig (fmaf(inp, wi1[l], fmaf(hl, wh1[l], b1[l])));
          float n   = ftanh_(fmaf(r, fmaf(hl, wh2[l], bh2[l]),
                                  fmaf(inp, wi2[l], bi2[l])));
          hl        = fmaf(z, hl - n, n);   // (1-z)*n + z*h
          h[l] = hl;
          inp  = hl;                        // feed next layer
        }
        my[t] = inp;                        // top-layer out -> next pass input
      }
    }

    // out[l, e] : [L, B, 1, N, 1] flattened; coalesced per layer
#pragma unroll
    for (int l = 0; l < LNUM; ++l) out[l * E + e] = h[l];
  }
}

extern "C" void kernel_launch(void* const* d_in, const int* in_sizes, int n_in,
                              void* d_out, int out_size, void* d_ws, size_t ws_size,
                              hipStream_t stream) {
  const float* x    = (const float*)d_in[0];
  const float* w_ih = (const float*)d_in[1];
  const float* w_hh = (const float*)d_in[2];
  const float* b_ih = (const float*)d_in[3];
  const float* b_hh = (const float*)d_in[4];
  float* out = (float*)d_out;

  const int E    = in_sizes[0] / TNUM;      // B*N = 65536 independent streams
  const int grid = (E + BLK - 1) / BLK;     // 256 blocks x 256 threads (8 waves)
  gru_stack_kernel<<<grid, BLK, 0, stream>>>(x, w_ih, w_hh, b_ih, b_hh, out, E);
}